// SpatialEncoder_60095182405927
// MI455X (gfx1250) — compile-verified
//
#include <hip/hip_runtime.h>

// ---------------------------------------------------------------------------
// SpatialEncoder on CDNA5 (gfx1250): all matmul-shaped ops run on the WMMA
// pipe via V_WMMA_F32_16X16X4_F32 (full fp32, wave32, 16x16 tiles).
// Heavy kernels block 2 M-tiles per wave so each activation (B) fragment
// feeds two WMMAs, halving the dominant gather/decode cost per FLOP.
// ---------------------------------------------------------------------------

typedef __attribute__((ext_vector_type(2))) float v2f;
typedef __attribute__((ext_vector_type(8))) float v8f;

#define BATCH 128
#define ENT   512
#define WMAP  64

__device__ __forceinline__ v8f wmma4(v2f a, v2f b, v8f c) {
  // D = A(16x4 f32) * B(4x16 f32) + C(16x16 f32)
  return __builtin_amdgcn_wmma_f32_16x16x4_f32(
      /*neg_a=*/false, a, /*neg_b=*/false, b,
      /*c_mod=*/(short)0, c, /*reuse_a=*/false, /*reuse_b=*/false);
}

// ---------------------------------------------------------------------------
__global__ void k_zero(float* __restrict__ p, long n) {
  long i = (long)blockIdx.x * blockDim.x + threadIdx.x;
  long stride = (long)gridDim.x * blockDim.x;
  for (; i < n; i += stride) p[i] = 0.0f;
}

// ---------------------------------------------------------------------------
// reduced = relu(emb[B,E,256] @ conv1_w[32,256]^T), scattered into
// sm[B,32,64,64] at per-entity flat index decoded from bit-packed coords.
// One wave = 16 entities x 32 out channels, K=256.
__global__ void k_conv1_scatter(const float* __restrict__ emb,
                                const int* __restrict__ exy,
                                const float* __restrict__ w,   // [32,256]
                                float* __restrict__ sm) {      // [B,32,4096]
  int wave = blockIdx.x * (blockDim.x >> 5) + (threadIdx.x >> 5);
  int lane = threadIdx.x & 31, lid = lane & 15, half = lane >> 4;
  int koff = half * 2;
  int b  = wave >> 5;   // E/16 = 32 entity tiles per batch
  int et = wave & 31;

  const float* aptr  = emb + ((size_t)b * ENT + (size_t)et * 16 + lid) * 256;
  const float* b0ptr = w + (size_t)lid * 256;          // out ch lid
  const float* b1ptr = w + (size_t)(16 + lid) * 256;   // out ch 16+lid

  v8f acc0 = {}, acc1 = {};
  for (int k = 0; k < 256; k += 4) {
    v2f a   = *(const v2f*)(aptr  + k + koff);
    v2f bb0 = *(const v2f*)(b0ptr + k + koff);
    v2f bb1 = *(const v2f*)(b1ptr + k + koff);
    acc0 = wmma4(a, bb0, acc0);
    acc1 = wmma4(a, bb1, acc1);
  }

  // decode big-endian 8-bit x/y for entity (et*16 + lid)
  const int* bp = exy + ((size_t)b * ENT + (size_t)et * 16 + lid) * 16;
  int xc = 0, yc = 0;
#pragma unroll
  for (int i = 0; i < 8; ++i) { xc = (xc << 1) | bp[i]; yc = (yc << 1) | bp[8 + i]; }
  int idx = yc * WMAP + xc;

  float* smb = sm + (size_t)b * 32 * 4096;
#pragma unroll
  for (int r = 0; r < 8; ++r) {
    int erow = half * 8 + r;               // entity row of this accumulator slot
    int idxr = __shfl(idx, erow, 32);
    float v0 = fmaxf(acc0[r], 0.0f);
    float v1 = fmaxf(acc1[r], 0.0f);
    smb[(size_t)lid * 4096 + idxr]        = v0;  // channel lid
    smb[(size_t)(16 + lid) * 4096 + idxr] = v1;  // channel 16+lid
  }
}

// ---------------------------------------------------------------------------
// 1x1 conv over concat([sm(32ch), x(24ch)]) -> 32ch, K=56, +bias, relu.
// One wave = 32 out channels x 16 positions.
__global__ void k_project(const float* __restrict__ sm,   // [B,32,4096]
                          const float* __restrict__ x,    // [B,24,4096]
                          const float* __restrict__ w,    // [32,56]
                          const float* __restrict__ bias, // [32]
                          float* __restrict__ out) {      // [B,32,4096]
  int wave = blockIdx.x * (blockDim.x >> 5) + (threadIdx.x >> 5);
  int lane = threadIdx.x & 31, lid = lane & 15, half = lane >> 4;
  int koff = half * 2;
  int b  = wave >> 8;    // 256 position tiles per batch
  int pt = wave & 255;
  int p  = pt * 16 + lid;

  const float* a0 = w + (size_t)lid * 56;
  const float* a1 = w + (size_t)(16 + lid) * 56;

  v8f acc0 = {}, acc1 = {};
  for (int k = 0; k < 56; k += 4) {
    v2f av0 = *(const v2f*)(a0 + k + koff);
    v2f av1 = *(const v2f*)(a1 + k + koff);
    v2f bv;
#pragma unroll
    for (int j = 0; j < 2; ++j) {
      int c = k + koff + j;
      const float* src = (c < 32) ? (sm + (((size_t)b * 32 + c) << 12))
                                  : (x  + (((size_t)b * 24 + (c - 32)) << 12));
      bv[j] = src[p];
    }
    acc0 = wmma4(av0, bv, acc0);
    acc1 = wmma4(av1, bv, acc1);
  }

  float* ob = out + (size_t)b * 32 * 4096;
#pragma unroll
  for (int r = 0; r < 8; ++r) {
    int oc = half * 8 + r;
    float v0 = fmaxf(acc0[r] + bias[oc], 0.0f);
    float v1 = fmaxf(acc1[r] + bias[16 + oc], 0.0f);
    ob[(size_t)oc * 4096 + p]        = v0;
    ob[(size_t)(16 + oc) * 4096 + p] = v1;
  }
}

// ---------------------------------------------------------------------------
// 4x4 stride-2 pad-1 conv, implicit GEMM. One wave = 32 out channels (2
// M-tiles sharing the B fragment) x 16 out positions, K = Cin*16.
// K-loop unrolled x4 so (ky,kx) and padding predicates become constants.
__global__ void k_conv4x4(const float* __restrict__ in, const float* __restrict__ w,
                          const float* __restrict__ bias, float* __restrict__ out,
                          int Cin, int Cout, int winShift, int woutShift,
                          int ntiles, int mtiles /* Cout/32 */) {
  int wave = blockIdx.x * (blockDim.x >> 5) + (threadIdx.x >> 5);
  int lane = threadIdx.x & 31, lid = lane & 15, half = lane >> 4;
  int koff = half * 2;
  int wpb = mtiles * ntiles;
  int b = wave / wpb, rest = wave % wpb;
  int mt = rest / ntiles, nt = rest % ntiles;

  int Win  = 1 << winShift;
  int Wout = 1 << woutShift;
  int pn = nt * 16 + lid;
  int oy = pn >> woutShift, ox = pn & (Wout - 1);
  int lda = Cin * 16;

  const float* a0 = w + (size_t)(mt * 32 + lid) * lda;
  const float* a1 = a0 + (size_t)16 * lda;
  const float* inb = in + ((size_t)b * Cin << (2 * winShift));

  v8f acc0 = {}, acc1 = {};
#pragma unroll 4
  for (int k = 0; k < lda; k += 4) {
    v2f av0 = *(const v2f*)(a0 + k + koff);
    v2f av1 = *(const v2f*)(a1 + k + koff);
    v2f bv;
#pragma unroll
    for (int j = 0; j < 2; ++j) {
      int kk = k + koff + j;
      int c = kk >> 4, rr = kk & 15, ky = rr >> 2, kx = rr & 3;
      int iy = 2 * oy - 1 + ky, ix = 2 * ox - 1 + kx;
      bool ok = (iy >= 0) & (iy < Win) & (ix >= 0) & (ix < Win);
      bv[j] = ok ? inb[((size_t)c << (2 * winShift)) + ((size_t)iy << winShift) + ix]
                 : 0.0f;
    }
    acc0 = wmma4(av0, bv, acc0);
    acc1 = wmma4(av1, bv, acc1);
  }

  float* ob = out + (((size_t)b * Cout + mt * 32) << (2 * woutShift));
#pragma unroll
  for (int r = 0; r < 8; ++r) {
    int oc = half * 8 + r;
    float v0 = fmaxf(acc0[r] + bias[mt * 32 + oc], 0.0f);
    float v1 = fmaxf(acc1[r] + bias[mt * 32 + 16 + oc], 0.0f);
    ob[((size_t)oc << (2 * woutShift)) + pn]        = v0;
    ob[((size_t)(16 + oc) << (2 * woutShift)) + pn] = v1;
  }
}

// ---------------------------------------------------------------------------
// 3x3 pad-1 conv at 8x8, C=128 (residual blocks). K=1152. One wave = 32 out
// channels (2 M-tiles sharing B) x 16 positions. Optional relu on input load
// (pre-activation), optional relu on output, optional skip add.
__global__ void k_conv3x3_res(const float* __restrict__ in, const float* __restrict__ w,
                              const float* __restrict__ bias,
                              const float* __restrict__ skip,
                              float* __restrict__ out, int reluIn, int reluOut) {
  int wave = blockIdx.x * (blockDim.x >> 5) + (threadIdx.x >> 5);
  int lane = threadIdx.x & 31, lid = lane & 15, half = lane >> 4;
  int koff = half * 2;
  int b = wave >> 4, rest = wave & 15;   // 4 m-tiles * 4 n-tiles per batch
  int mt = rest >> 2, nt = rest & 3;
  int pn = nt * 16 + lid, oy = pn >> 3, ox = pn & 7;
  const int lda = 128 * 9;

  const float* a0 = w + (size_t)(mt * 32 + lid) * lda;
  const float* a1 = a0 + (size_t)16 * lda;
  const float* inb = in + ((size_t)b << 13);   // 128*64

  v8f acc0 = {}, acc1 = {};
#pragma unroll 2
  for (int k = 0; k < lda; k += 4) {
    v2f av0 = *(const v2f*)(a0 + k + koff);
    v2f av1 = *(const v2f*)(a1 + k + koff);
    v2f bv;
#pragma unroll
    for (int j = 0; j < 2; ++j) {
      int kk = k + koff + j;
      int c = kk / 9, rr = kk - c * 9, ky = rr / 3, kx = rr - ky * 3;
      int iy = oy - 1 + ky, ix = ox - 1 + kx;
      bool ok = (iy >= 0) & (iy < 8) & (ix >= 0) & (ix < 8);
      float v = ok ? inb[((size_t)c << 6) + (iy << 3) + ix] : 0.0f;
      if (reluIn) v = fmaxf(v, 0.0f);
      bv[j] = v;
    }
    acc0 = wmma4(av0, bv, acc0);
    acc1 = wmma4(av1, bv, acc1);
  }

  size_t obase = ((size_t)b << 13) + ((size_t)(mt * 32) << 6);
#pragma unroll
  for (int r = 0; r < 8; ++r) {
    int oc = half * 8 + r;
    float v0 = acc0[r] + bias[mt * 32 + oc];
    float v1 = acc1[r] + bias[mt * 32 + 16 + oc];
    if (reluOut) { v0 = fmaxf(v0, 0.0f); v1 = fmaxf(v1, 0.0f); }
    size_t o0 = obase + ((size_t)oc << 6) + pn;
    size_t o1 = obase + ((size_t)(16 + oc) << 6) + pn;
    if (skip) { v0 += skip[o0]; v1 += skip[o1]; }
    out[o0] = v0;
    out[o1] = v1;
  }
}

// ---------------------------------------------------------------------------
// FC: out[128,256] = relu(relu(h[128,8192]) @ fc_w[256,8192]^T + fc_b)
// One wave = 32 batch rows (2 M-tiles sharing the weight fragment) x 16 outs.
__global__ void k_fc(const float* __restrict__ h, const float* __restrict__ w,
                     const float* __restrict__ bias, float* __restrict__ out) {
  int wave = blockIdx.x * (blockDim.x >> 5) + (threadIdx.x >> 5);
  int lane = threadIdx.x & 31, lid = lane & 15, half = lane >> 4;
  int koff = half * 2;
  int mt = wave >> 4, nt = wave & 15;   // 4 batch tiles (x32) x 16 out tiles

  const float* a0 = h + ((size_t)(mt * 32 + lid) << 13);
  const float* a1 = a0 + ((size_t)16 << 13);
  const float* bp = w + ((size_t)(nt * 16 + lid) << 13);

  v8f acc0 = {}, acc1 = {};
  for (int k = 0; k < 8192; k += 4) {
    v2f av0 = *(const v2f*)(a0 + k + koff);
    v2f av1 = *(const v2f*)(a1 + k + koff);
    av0[0] = fmaxf(av0[0], 0.0f); av0[1] = fmaxf(av0[1], 0.0f);
    av1[0] = fmaxf(av1[0], 0.0f); av1[1] = fmaxf(av1[1], 0.0f);
    v2f bv = *(const v2f*)(bp + k + koff);
    acc0 = wmma4(av0, bv, acc0);
    acc1 = wmma4(av1, bv, acc1);
  }

  float bb = bias[nt * 16 + lid];
#pragma unroll
  for (int r = 0; r < 8; ++r) {
    int bm0 = mt * 32 + half * 8 + r;
    out[(size_t)bm0 * 256 + nt * 16 + lid]        = fmaxf(acc0[r] + bb, 0.0f);
    out[(size_t)(bm0 + 16) * 256 + nt * 16 + lid] = fmaxf(acc1[r] + bb, 0.0f);
  }
}

// ---------------------------------------------------------------------------
extern "C" void kernel_launch(void* const* d_in, const int* in_sizes, int n_in,
                              void* d_out, int out_size, void* d_ws, size_t ws_size,
                              hipStream_t stream) {
  (void)in_sizes; (void)n_in; (void)out_size; (void)ws_size;

  const float* x   = (const float*)d_in[0];
  const float* emb = (const float*)d_in[1];
  const int*   exy = (const int*)d_in[2];
  const float* c1w = (const float*)d_in[3];
  const float* psw = (const float*)d_in[4];
  const float* psb = (const float*)d_in[5];
  const float* d1w = (const float*)d_in[6];
  const float* d1b = (const float*)d_in[7];
  const float* d2w = (const float*)d_in[8];
  const float* d2b = (const float*)d_in[9];
  const float* d3w = (const float*)d_in[10];
  const float* d3b = (const float*)d_in[11];
  const float* rw1 = (const float*)d_in[12];
  const float* rb1 = (const float*)d_in[13];
  const float* rw2 = (const float*)d_in[14];
  const float* rb2 = (const float*)d_in[15];
  const float* fcw = (const float*)d_in[16];
  const float* fcb = (const float*)d_in[17];
  float* out = (float*)d_out;
  float* ws  = (float*)d_ws;

  // workspace layout (floats), with reuse:
  //   [0 .. 16M)   : scatter map [B,32,64,64]  -> later h1 [B,64,32,32]
  //   [16M .. 32M) : h0 [B,32,64,64]           -> later h2 [B,128,16,16] + tmp
  float* sm  = ws;
  float* h0  = ws + 16777216;
  float* h1  = ws;                        // reuse (8,388,608 floats)
  float* h2  = ws + 16777216;             // reuse (4,194,304 floats)
  float* tmp = ws + 16777216 + 2097152;   // res temp (1,048,576 floats)
  const size_t SL = 1048576;              // one map_skip slice [B,128,8,8]
  float* emb_out = out + 5 * SL;

  dim3 blk(128);  // 4 waves per block

  k_zero<<<2048, 256, 0, stream>>>(sm, 16777216L);
  k_conv1_scatter<<<1024, blk, 0, stream>>>(emb, exy, c1w, sm);
  k_project<<<8192, blk, 0, stream>>>(sm, x, psw, psb, h0);
  // ds1: 32->64, 64x64 -> 32x32   (ntiles=64, mtiles=2)  16384 waves
  k_conv4x4<<<4096, blk, 0, stream>>>(h0, d1w, d1b, h1, 32, 64, 6, 5, 64, 2);
  // ds2: 64->128, 32x32 -> 16x16  (ntiles=16, mtiles=4)  8192 waves
  k_conv4x4<<<2048, blk, 0, stream>>>(h1, d2w, d2b, h2, 64, 128, 5, 4, 16, 4);
  // ds3: 128->128, 16x16 -> 8x8   (ntiles=4, mtiles=4)   2048 waves -> map_skip[0]
  k_conv4x4<<<512, blk, 0, stream>>>(h2, d3w, d3b, out, 128, 128, 4, 3, 4, 4);

  for (int i = 0; i < 4; ++i) {
    const float* hin = out + (size_t)i * SL;
    float* hout = out + (size_t)(i + 1) * SL;
    // t = relu(conv1(relu(h)) + b1)
    k_conv3x3_res<<<512, blk, 0, stream>>>(hin, rw1 + (size_t)i * 147456,
                                           rb1 + i * 128, nullptr, tmp, 1, 1);
    // h = conv2(t) + b2 + h
    k_conv3x3_res<<<512, blk, 0, stream>>>(tmp, rw2 + (size_t)i * 147456,
                                           rb2 + i * 128, hin, hout, 0, 0);
  }

  k_fc<<<16, blk, 0, stream>>>(out + 4 * SL, fcw, fcb, emb_out);
}